// KuramotoLayer_76991583748503
// MI455X (gfx1250) — compile-verified
//
#include <hip/hip_runtime.h>
#include <math.h>

#define N        4096
#define N_STEPS  50
#define DT       0.01f

typedef float v2f __attribute__((ext_vector_type(2)));
typedef float v8f __attribute__((ext_vector_type(8)));

// --- per-step sin/cos of theta (tiny, 4096 lanes) -------------------------
__global__ void kuramoto_sincos_kernel(const float* __restrict__ theta,
                                       float* __restrict__ s,
                                       float* __restrict__ c) {
  int i = blockIdx.x * blockDim.x + threadIdx.x;
  if (i < N) {
    float t = theta[i];
    s[i] = sinf(t);
    c[i] = cosf(t);
  }
}

// --- one Euler step: theta_out = theta_in + dt*(omega + cos*(K s) - sin*(K c))
//
// Block = 256 threads = 8 waves, owns 16 output rows (row0..row0+15).
// Wave w accumulates k in [w*512, (w+1)*512) with V_WMMA_F32_16X16X4_F32:
//   A (16x4):  row 0 = s[k..k+3], row 1 = c[k..k+3], rows 2..15 = 0
//   B (4x16):  K[k..k+3][row0..row0+15]   (coalesced: lane n -> column row0+n)
//   D row 0  = partial (K·s)[row0+n],  D row 1 = partial (K·c)[row0+n]
// (valid because K is symmetric: K·v == Kᵀ·v)
//
// A-fragment selection (s for row 0, c for row 1, zero otherwise) is resolved
// ONCE per lane as a base index into a combined LDS array [s | c | zeros], so
// the inner loop is branch-free: 1 ds_load_b64 + 2 global_load_b32 per WMMA,
// with a compile-time-constant trip count (scalar loop control, EXEC stays all-1s).
__global__ __launch_bounds__(256)
void kuramoto_step_kernel(const float* __restrict__ Kmat,
                          const float* __restrict__ s_g,
                          const float* __restrict__ c_g,
                          const float* __restrict__ omegas,
                          const float* __restrict__ theta_in,
                          float* __restrict__ theta_out) {
  __shared__ float lds_v[3 * N];     // [0,N): sin, [N,2N): cos, [2N,3N): zeros
  __shared__ float lds_ks[8 * 16];
  __shared__ float lds_kc[8 * 16];

  const int tid  = threadIdx.x;
  const int row0 = blockIdx.x * 16;

  // Stage sin/cos vectors + zero page into LDS (coalesced float4).
  for (int i = tid; i < N / 4; i += 256) {
    ((float4*)lds_v)[i]             = ((const float4*)s_g)[i];
    ((float4*)lds_v)[N / 4 + i]     = ((const float4*)c_g)[i];
    ((float4*)lds_v)[2 * N / 4 + i] = make_float4(0.f, 0.f, 0.f, 0.f);
  }
  __syncthreads();

  const int wave  = tid >> 5;
  const int lane  = tid & 31;
  const int lm    = lane & 15;          // N index within fragment
  const int khalf = (lane >> 4) << 1;   // lanes 0-15: k+0/k+1, lanes 16-31: k+2/k+3
  const int kbase = wave * (N / 8);     // 512 k-values per wave

  // Per-lane A source resolved once: row 0 -> sin, row 1 -> cos, rows 2..15 -> zeros.
  const int abase = ((lm == 0) ? 0 : ((lm == 1) ? N : 2 * N)) + kbase + khalf;

  // B source: column row0+lm of K, starting at row kbase+khalf.
  const float* bptr = Kmat + (size_t)(kbase + khalf) * N + row0 + lm;

  v8f acc0 = {};
  v8f acc1 = {};

#pragma unroll 4
  for (int it = 0; it < (N / 8) / 8; ++it) {   // 64 iterations, 2 WMMAs each
    const int ai = abase + it * 8;

    v2f a0 = *(const v2f*)(&lds_v[ai]);        // ds_load_b64 (8B aligned)
    v2f a1 = *(const v2f*)(&lds_v[ai + 4]);

    const float* bp = bptr + (size_t)it * 8 * N;
    v2f b0, b1;
    b0.x = bp[0];                              // K[kk  ][row0+lm]
    b0.y = bp[(size_t)N];                      // K[kk+1][row0+lm]
    b1.x = bp[(size_t)4 * N];                  // K[kk+4][row0+lm]
    b1.y = bp[(size_t)5 * N];                  // K[kk+5][row0+lm]

    acc0 = __builtin_amdgcn_wmma_f32_16x16x4_f32(false, a0, false, b0, (short)0,
                                                 acc0, false, false);
    acc1 = __builtin_amdgcn_wmma_f32_16x16x4_f32(false, a1, false, b1, (short)0,
                                                 acc1, false, false);
  }

  // D[0,n] (lanes 0-15) = partial (K s)[row0+n]; D[1,n] = partial (K c)[row0+n]
  const float ks_p = acc0[0] + acc1[0];
  const float kc_p = acc0[1] + acc1[1];
  if (lane < 16) {
    lds_ks[wave * 16 + lane] = ks_p;
    lds_kc[wave * 16 + lane] = kc_p;
  }
  __syncthreads();

  if (tid < 16) {
    float ks = 0.0f, kc = 0.0f;
#pragma unroll
    for (int w = 0; w < 8; ++w) {
      ks += lds_ks[w * 16 + tid];
      kc += lds_kc[w * 16 + tid];
    }
    const int   i  = row0 + tid;
    const float si = lds_v[i];          // sin(theta_i)
    const float ci = lds_v[N + i];      // cos(theta_i)
    const float coupling = ci * ks - si * kc;  // sum_j K[i,j]*sin(theta_j - theta_i)
    theta_out[i] = theta_in[i] + DT * (omegas[i] + coupling);
  }
}

extern "C" void kernel_launch(void* const* d_in, const int* in_sizes, int n_in,
                              void* d_out, int out_size, void* d_ws, size_t ws_size,
                              hipStream_t stream) {
  const float* phases = (const float*)d_in[0];   // (N,)
  const float* Kmat   = (const float*)d_in[1];   // (N,N) symmetric
  const float* omegas = (const float*)d_in[2];   // (N,)
  float*       out    = (float*)d_out;           // (N,)

  float* ws  = (float*)d_ws;                     // 4*N floats = 64 KB used
  float* thA = ws;
  float* thB = ws + N;
  float* sv  = ws + 2 * N;
  float* cv  = ws + 3 * N;

  const float* cur = phases;
  for (int t = 0; t < N_STEPS; ++t) {
    float* nxt = (t == N_STEPS - 1) ? out : ((t & 1) ? thB : thA);
    kuramoto_sincos_kernel<<<N / 256, 256, 0, stream>>>(cur, sv, cv);
    kuramoto_step_kernel<<<N / 16, 256, 0, stream>>>(Kmat, sv, cv, omegas, cur, nxt);
    cur = nxt;
  }
}